// MatsimGNN_80307298501313
// MI455X (gfx1250) — compile-verified
//
#include <hip/hip_runtime.h>
#include <hip/hip_bf16.h>

typedef __attribute__((ext_vector_type(2))) float v2f;
typedef __attribute__((ext_vector_type(8))) float v8f;

#define THREADS 256

static __host__ unsigned nblocks(size_t n) { return (unsigned)((n + THREADS - 1) / THREADS); }

// ---------------- utility kernels ----------------

__global__ void fill_f32_kernel(float* __restrict__ p, float v, size_t n) {
    size_t i = (size_t)blockIdx.x * blockDim.x + threadIdx.x;
    if (i < n) p[i] = v;
}

__global__ void deg_scatter_kernel(const int* __restrict__ dst, float* __restrict__ deg, int E) {
    int e = blockIdx.x * blockDim.x + threadIdx.x;
    if (e < E) atomicAdd(&deg[dst[e]], 1.0f);
}

// in-place: deg -> dis (rsqrt), and selfc = 1/deg
__global__ void dis_kernel(float* __restrict__ deg_dis, float* __restrict__ selfc, int n) {
    int i = blockIdx.x * blockDim.x + threadIdx.x;
    if (i < n) {
        float d = deg_dis[i];
        deg_dis[i] = rsqrtf(d);
        selfc[i]   = 1.0f / d;   // dis*dis
    }
}

__global__ void norm_kernel(const int* __restrict__ src, const int* __restrict__ dst,
                            const float* __restrict__ dis, float* __restrict__ norm, int E) {
    int e = blockIdx.x * blockDim.x + threadIdx.x;
    if (e < E) norm[e] = dis[src[e]] * dis[dst[e]];
}

// ---------------- WMMA fp32 GEMM: C[N,Cout] = A[N,K] @ B[K,Cout] ----------------
// One wave computes one 16x16 output tile using V_WMMA_F32_16X16X4_F32.
// Requires: Nrows % 16 == 0, K % 4 == 0, Cout % 16 == 0 (true here: 100000, {64,128}, {64,128,256}).

__global__ void gemm_wmma_f32_kernel(const float* __restrict__ A,
                                     const float* __restrict__ B,
                                     float* __restrict__ C,
                                     int Nrows, int K, int Cout) {
    const int lane  = threadIdx.x & 31;
    const int wave  = (int)((blockIdx.x * blockDim.x + threadIdx.x) >> 5);
    const int ntc   = Cout >> 4;                      // tiles along Cout
    const int tiles = (Nrows >> 4) * ntc;
    if (wave >= tiles) return;                        // wave-uniform: EXEC stays all-1s

    const int tileM = wave / ntc;
    const int tileN = wave % ntc;

    const int row   = tileM * 16 + (lane & 15);       // A row (M) for this lane
    const int col   = tileN * 16 + (lane & 15);       // B/D column (N) for this lane
    const int khalf = (lane >> 4) * 2;                // lanes 0-15: K=0,1 ; lanes 16-31: K=2,3

    const float* arow = A + (size_t)row * K;

    v8f acc = {};
    for (int k = 0; k < K; k += 4) {
        v2f a, b;
        a.x = arow[k + khalf];
        a.y = arow[k + khalf + 1];
        b.x = B[(size_t)(k + khalf) * Cout + col];
        b.y = B[(size_t)(k + khalf + 1) * Cout + col];
        acc = __builtin_amdgcn_wmma_f32_16x16x4_f32(
            /*neg_a=*/false, a, /*neg_b=*/false, b,
            /*c_mod=*/(short)0, acc, /*reuse_a=*/false, /*reuse_b=*/false);
    }

    // D layout: lanes 0-15 hold rows M=0..7 in vgpr 0..7 (N=lane); lanes 16-31 rows M=8..15.
    const int base_m = tileM * 16 + (lane >> 4) * 8;
#pragma unroll
    for (int i = 0; i < 8; ++i) {
        C[(size_t)(base_m + i) * Cout + col] = acc[i];
    }
}

// ---------------- GCN aggregation ----------------

// agg[i,c] = h[i,c] * selfc[i] + bias[c]     (vectorized float4)
__global__ void agg_init_kernel(const float* __restrict__ h, const float* __restrict__ selfc,
                                const float* __restrict__ bias, float* __restrict__ agg,
                                int n, int C) {
    const int cq = C >> 2;
    size_t idx = (size_t)blockIdx.x * blockDim.x + threadIdx.x;
    size_t total = (size_t)n * cq;
    if (idx >= total) return;
    int i = (int)(idx / cq);
    int c = (int)(idx % cq) * 4;
    float s = selfc[i];
    float4 hv = *(const float4*)(h + (size_t)i * C + c);
    float4 bv = *(const float4*)(bias + c);
    float4 r;
    r.x = hv.x * s + bv.x;
    r.y = hv.y * s + bv.y;
    r.z = hv.z * s + bv.z;
    r.w = hv.w * s + bv.w;
    *(float4*)(agg + (size_t)i * C + c) = r;
}

// agg[dst[e], :] += h[src[e], :] * norm[e]   (one thread per edge x 4-channel quad)
__global__ void edge_scatter_kernel(const int* __restrict__ src, const int* __restrict__ dst,
                                    const float* __restrict__ norm, const float* __restrict__ h,
                                    float* __restrict__ agg, int E, int C) {
    const int cq = C >> 2;
    size_t idx = (size_t)blockIdx.x * blockDim.x + threadIdx.x;
    size_t total = (size_t)E * cq;
    if (idx >= total) return;
    int e = (int)(idx / cq);
    int c = (int)(idx % cq) * 4;
    int s = src[e];
    int d = dst[e];
    float w = norm[e];
    float4 hv = *(const float4*)(h + (size_t)s * C + c);
    float* out = agg + (size_t)d * C + c;
    atomicAdd(out + 0, hv.x * w);
    atomicAdd(out + 1, hv.y * w);
    atomicAdd(out + 2, hv.z * w);
    atomicAdd(out + 3, hv.w * w);
}

__global__ void relu_kernel(float* __restrict__ p, size_t n4) {
    size_t idx = (size_t)blockIdx.x * blockDim.x + threadIdx.x;
    if (idx >= n4) return;
    float4 v = *(float4*)(p + idx * 4);
    v.x = fmaxf(v.x, 0.0f);
    v.y = fmaxf(v.y, 0.0f);
    v.z = fmaxf(v.z, 0.0f);
    v.w = fmaxf(v.w, 0.0f);
    *(float4*)(p + idx * 4) = v;
}

// ---------------- pooling + head ----------------

#define NGRAPHS 8

// pooled[g,c] += sum over nodes i with batch[i]==g of h[i,c]; counts[g] += #nodes
__global__ void pool_sum_kernel(const float* __restrict__ h, const int* __restrict__ batch,
                                float* __restrict__ pooled, float* __restrict__ counts,
                                int n, int C) {
    extern __shared__ float smem[];           // NGRAPHS*C + NGRAPHS
    const int nbins = NGRAPHS * C;
    for (int i = threadIdx.x; i < nbins + NGRAPHS; i += blockDim.x) smem[i] = 0.0f;
    __syncthreads();

    size_t total  = (size_t)n * C;
    size_t stride = (size_t)gridDim.x * blockDim.x;
    for (size_t idx = (size_t)blockIdx.x * blockDim.x + threadIdx.x; idx < total; idx += stride) {
        int i = (int)(idx / C);
        int c = (int)(idx % C);
        int g = batch[i];
        atomicAdd(&smem[g * C + c], h[idx]);
        if (c == 0) atomicAdd(&smem[nbins + g], 1.0f);
    }
    __syncthreads();

    for (int i = threadIdx.x; i < nbins; i += blockDim.x) {
        float v = smem[i];
        if (v != 0.0f) atomicAdd(&pooled[i], v);
    }
    if (threadIdx.x < NGRAPHS) {
        float v = smem[nbins + threadIdx.x];
        if (v != 0.0f) atomicAdd(&counts[threadIdx.x], v);
    }
}

// out[g] = (sum_c pooled[g,c] * Wfc[c]) / counts[g] + bfc[0]
__global__ void final_fc_kernel(const float* __restrict__ pooled, const float* __restrict__ counts,
                                const float* __restrict__ Wfc, const float* __restrict__ bfc,
                                float* __restrict__ out, int C) {
    int g = threadIdx.x;
    if (g < NGRAPHS) {
        float s = 0.0f;
        const float* row = pooled + (size_t)g * C;
        for (int c = 0; c < C; ++c) s += row[c] * Wfc[c];
        out[g] = s / counts[g] + bfc[0];
    }
}

// ---------------- launch ----------------

extern "C" void kernel_launch(void* const* d_in, const int* in_sizes, int n_in,
                              void* d_out, int out_size, void* d_ws, size_t ws_size,
                              hipStream_t stream) {
    const float* x    = (const float*)d_in[0];
    const int*   edge = (const int*)d_in[1];
    const int*   batch= (const int*)d_in[2];
    const float* W1   = (const float*)d_in[3];
    const float* b1   = (const float*)d_in[4];
    const float* W2   = (const float*)d_in[5];
    const float* b2   = (const float*)d_in[6];
    const float* W3   = (const float*)d_in[7];
    const float* b3   = (const float*)d_in[8];
    const float* Wfc  = (const float*)d_in[9];
    const float* bfc  = (const float*)d_in[10];
    float* out = (float*)d_out;

    const int C0 = 128;
    const int N  = in_sizes[0] / C0;
    const int E  = in_sizes[1] / 2;
    const int* src = edge;
    const int* dst = edge + E;

    // workspace layout (256B-aligned slices)
    auto align_up = [](size_t v) { return (v + 255) & ~(size_t)255; };
    char* w = (char*)d_ws;
    float* deg_dis = (float*)w;  w += align_up((size_t)N * 4);          // deg, then dis (in place)
    float* selfc   = (float*)w;  w += align_up((size_t)N * 4);
    float* norm    = (float*)w;  w += align_up((size_t)E * 4);
    float* bufH    = (float*)w;  w += align_up((size_t)N * 256 * 4);
    float* bufAgg  = (float*)w;  w += align_up((size_t)N * 256 * 4);
    float* pooled  = (float*)w;  w += align_up((size_t)NGRAPHS * 256 * 4);
    float* counts  = (float*)w;  w += align_up((size_t)NGRAPHS * 4);
    (void)ws_size; (void)n_in; (void)out_size;

    // --- degree / normalization ---
    fill_f32_kernel<<<nblocks(N), THREADS, 0, stream>>>(deg_dis, 1.0f, (size_t)N);   // self-loop
    fill_f32_kernel<<<nblocks(NGRAPHS * 256), THREADS, 0, stream>>>(pooled, 0.0f, (size_t)NGRAPHS * 256);
    fill_f32_kernel<<<1, THREADS, 0, stream>>>(counts, 0.0f, (size_t)NGRAPHS);
    deg_scatter_kernel<<<nblocks(E), THREADS, 0, stream>>>(dst, deg_dis, E);
    dis_kernel<<<nblocks(N), THREADS, 0, stream>>>(deg_dis, selfc, N);
    norm_kernel<<<nblocks(E), THREADS, 0, stream>>>(src, dst, deg_dis, norm, E);

    const float* h_in = x;
    const float* Ws[3] = {W1, W2, W3};
    const float* bs[3] = {b1, b2, b3};
    const int cin[3]  = {128, 64, 128};
    const int cout[3] = {64, 128, 256};

    for (int L = 0; L < 3; ++L) {
        const int K = cin[L], C = cout[L];
        // GEMM via WMMA: bufH = h_in @ W
        size_t waves = (size_t)(N / 16) * (C / 16);
        gemm_wmma_f32_kernel<<<nblocks(waves * 32), THREADS, 0, stream>>>(h_in, Ws[L], bufH, N, K, C);
        // agg = h*selfc + b
        agg_init_kernel<<<nblocks((size_t)N * (C / 4)), THREADS, 0, stream>>>(bufH, selfc, bs[L], bufAgg, N, C);
        // agg[dst] += h[src]*norm
        edge_scatter_kernel<<<nblocks((size_t)E * (C / 4)), THREADS, 0, stream>>>(src, dst, norm, bufH, bufAgg, E, C);
        // relu in place
        relu_kernel<<<nblocks((size_t)N * C / 4), THREADS, 0, stream>>>(bufAgg, (size_t)N * C / 4);
        h_in = bufAgg;
    }

    // --- pool + head ---
    size_t shmem = (size_t)(NGRAPHS * 256 + NGRAPHS) * sizeof(float);
    pool_sum_kernel<<<1024, THREADS, shmem, stream>>>(bufAgg, batch, pooled, counts, N, 256);
    final_fc_kernel<<<1, 32, 0, stream>>>(pooled, counts, Wfc, bfc, out, 256);
}